// PointSupervisionLoss_72705206386770
// MI455X (gfx1250) — compile-verified
//
#include <hip/hip_runtime.h>
#include <math.h>

typedef __attribute__((ext_vector_type(2))) float v2f;
typedef __attribute__((ext_vector_type(8))) float v8f;

namespace {
constexpr int kB = 8;
constexpr int kH = 512;
constexpr int kW = 512;
constexpr int kN = 50;
constexpr int kPix = kH * kW;                // 262144 pixels per batch
constexpr int kBlocksPerBatch = 256;         // 1024 pixels per block (8 waves * 128)
constexpr int kGrid = kB * kBlocksPerBatch;  // 2048 blocks
}

// V_WMMA_F32_16X16X4_F32 with POINTS on M (A) and PIXELS on N (B):
//   A row (point q):  [-2qy, -2qx, q^2, 0]
//   B col (pixel p):  [y, x, 1, 0]
//   => D[m][n] = q^2 - 2 p.q   (|p|^2 cancels in the pos-vs-neg comparison;
//      all values are integers < 2^24 so fp32 WMMA accumulation is exact).
// C/D layout: VGPR r, lanes 0-15 = M=r (N=lane); lanes 16-31 = M=r+8 (N=lane-16).
// => min over points = per-lane min across the 8 VGPRs + one xor-16 merge;
//    lane l (<16) then holds the min for pixel l directly.
__global__ __launch_bounds__(256) void psl_mask_loss_kernel(
    const float* __restrict__ x_in,   // (B,1,H,W) logits
    const int* __restrict__ posc,     // (B,N,2) int32
    const int* __restrict__ negc,     // (B,N,2) int32
    float* __restrict__ target,       // (B,1,H,W) mask out
    float* __restrict__ partial)      // per-block loss partials (kGrid floats)
{
  const int lane = threadIdx.x & 31;
  const int wave = threadIdx.x >> 5;
  const int b = blockIdx.x >> 8;  // 256 blocks per batch
  const int waveBase = (blockIdx.x & 255) * 1024 + wave * 128;
  const bool hi = lane >= 16;
  const int nl = lane & 15;

  // --- A tiles: 4 tiles x 16 points, pos and neg; pad n>=50 to +huge.
  // A 16x4 f32 layout (ISA table): lanes 0-15 hold K=0,1; lanes 16-31 hold K=2,3.
  v2f ap[4], an[4];
#pragma unroll
  for (int j = 0; j < 4; ++j) {
    const int n = j * 16 + nl;
    const int nc = (n < kN) ? n : 0;
    float py = (float)posc[(b * kN + nc) * 2 + 0];
    float px = (float)posc[(b * kN + nc) * 2 + 1];
    float ny = (float)negc[(b * kN + nc) * 2 + 0];
    float nx = (float)negc[(b * kN + nc) * 2 + 1];
    float pq2 = fmaf(py, py, px * px);
    float nq2 = fmaf(ny, ny, nx * nx);
    if (n >= kN) { py = 0.f; px = 0.f; ny = 0.f; nx = 0.f; pq2 = 3.0e38f; nq2 = 3.0e38f; }
    ap[j].x = hi ? pq2 : (-2.0f * py);
    ap[j].y = hi ? 0.f : (-2.0f * px);
    an[j].x = hi ? nq2 : (-2.0f * ny);
    an[j].y = hi ? 0.f : (-2.0f * nx);
  }

  float acc = 0.f;
#pragma unroll 1
  for (int t = 0; t < 8; ++t) {
    const int P0 = waveBase + t * 16;  // 16 consecutive pixels, same image row
    const int y = P0 >> 9;
    const int x0 = P0 & 511;
    // B 4x16 f32 (assumed mirror of A layout): lanes 0-15 = K0,K1 for N=lane;
    // lanes 16-31 = K2,K3 for N=lane-16.
    v2f bpix;
    bpix.x = hi ? 1.0f : (float)y;
    bpix.y = hi ? 0.0f : (float)(x0 + nl);

    const v8f cz = {};
    v8f cp = __builtin_amdgcn_wmma_f32_16x16x4_f32(false, ap[0], false, bpix, (short)0, cz, false, false);
    v8f cn = __builtin_amdgcn_wmma_f32_16x16x4_f32(false, an[0], false, bpix, (short)0, cz, false, false);
#pragma unroll
    for (int j = 1; j < 4; ++j) {
      v8f tp = __builtin_amdgcn_wmma_f32_16x16x4_f32(false, ap[j], false, bpix, (short)0, cz, false, false);
      v8f tn = __builtin_amdgcn_wmma_f32_16x16x4_f32(false, an[j], false, bpix, (short)0, cz, false, false);
#pragma unroll
      for (int i = 0; i < 8; ++i) {
        cp[i] = fminf(cp[i], tp[i]);
        cn[i] = fminf(cn[i], tn[i]);
      }
    }

    // Min over points: per-lane across the 8 accumulator VGPRs...
    float vp = cp[0], vn = cn[0];
#pragma unroll
    for (int i = 1; i < 8; ++i) {
      vp = fminf(vp, cp[i]);
      vn = fminf(vn, cn[i]);
    }
    // ...then merge M=0..7 (lanes 0-15) with M=8..15 (lanes 16-31).
    vp = fminf(vp, __shfl_xor(vp, 16, 32));
    vn = fminf(vn, __shfl_xor(vn, 16, 32));

    if (lane < 16) {
      const float z = (vp < vn) ? 1.0f : 0.0f;
      const int idx = b * kPix + y * kW + x0 + lane;
      target[idx] = z;
      const float xv = x_in[idx];
      acc += fmaxf(xv, 0.f) - xv * z + log1pf(expf(-fabsf(xv)));
    }
  }

  // Deterministic wave -> block reduction of the loss partial.
  acc += __shfl_xor(acc, 1, 32);
  acc += __shfl_xor(acc, 2, 32);
  acc += __shfl_xor(acc, 4, 32);
  acc += __shfl_xor(acc, 8, 32);
  acc += __shfl_xor(acc, 16, 32);
  __shared__ float sred[8];
  if (lane == 0) sred[wave] = acc;
  __syncthreads();
  if (threadIdx.x == 0) {
    float s = 0.f;
#pragma unroll
    for (int i = 0; i < 8; ++i) s += sred[i];
    partial[blockIdx.x] = s;
  }
}

// Fixed-order final reduction: deterministic across replays.
__global__ __launch_bounds__(256) void psl_reduce_kernel(
    const float* __restrict__ partial, float* __restrict__ loss) {
  __shared__ double sm[256];
  double s = 0.0;
  for (int i = threadIdx.x; i < kGrid; i += 256) s += (double)partial[i];
  sm[threadIdx.x] = s;
  __syncthreads();
  for (int off = 128; off > 0; off >>= 1) {
    if ((int)threadIdx.x < off) sm[threadIdx.x] += sm[threadIdx.x + off];
    __syncthreads();
  }
  if (threadIdx.x == 0) loss[0] = (float)(sm[0] / (double)(kB * kPix));
}

extern "C" void kernel_launch(void* const* d_in, const int* in_sizes, int n_in,
                              void* d_out, int out_size, void* d_ws, size_t ws_size,
                              hipStream_t stream) {
  (void)in_sizes; (void)n_in; (void)out_size; (void)ws_size;
  const float* x_in = (const float*)d_in[0];   // out: (8,1,512,512) f32
  const int* posc  = (const int*)d_in[1];      // pos_clicks: (8,50,2) i32
  const int* negc  = (const int*)d_in[2];      // neg_clicks: (8,50,2) i32
  float* loss = (float*)d_out;                 // [0] = loss
  float* target = loss + 1;                    // [1..] = pseudo_target flat
  float* partial = (float*)d_ws;               // kGrid floats of scratch

  psl_mask_loss_kernel<<<kGrid, 256, 0, stream>>>(x_in, posc, negc, target, partial);
  psl_reduce_kernel<<<1, 256, 0, stream>>>(partial, loss);
}